// GCMCGraphConv_23227183136841
// MI455X (gfx1250) — compile-verified
//
#include <hip/hip_runtime.h>

typedef float v2f __attribute__((ext_vector_type(2)));
typedef float v4f __attribute__((ext_vector_type(4)));
typedef float v8f __attribute__((ext_vector_type(8)));

#define D_FEAT 64
#define TILES_PER_WAVE 4   // 64 edges per wave
#define WAVES_PER_BLOCK 8  // 256 threads
#define ROW_PAD_F 68       // 64 floats + 4 pad -> 272B row stride (bank-conflict-free)
#define TILE_F (16 * ROW_PAD_F)

// ---------------------------------------------------------------------------
// Zero the accumulator (d_out is poisoned by the harness). b128 stores.
// ---------------------------------------------------------------------------
__global__ void __launch_bounds__(256) gcmc_zero_kernel(v4f* __restrict__ out, int n4) {
  int i = blockIdx.x * blockDim.x + threadIdx.x;
  if (i < n4) out[i] = (v4f){0.f, 0.f, 0.f, 0.f};
}

// ---------------------------------------------------------------------------
// Async-stage one 16-row review_feat tile (4KB) into LDS with 272B row pitch.
// 8 x global_load_async_to_lds_b128, each 512B coalesced. Tracked by ASYNCcnt.
// ---------------------------------------------------------------------------
__device__ __forceinline__ void stage_tile_fast(const float* __restrict__ rf,
                                                int tile, unsigned lds_base,
                                                int lane) {
  unsigned long long g = (unsigned long long)(uintptr_t)rf +
                         (unsigned long long)tile * 256ull + (unsigned)(lane * 16);
  unsigned l = lds_base + (unsigned)((lane >> 4) * 272 + (lane & 15) * 16);
#pragma unroll
  for (int i = 0; i < 8; ++i) {
    asm volatile("global_load_async_to_lds_b128 %0, %1, off"
                 :: "v"(l), "v"(g) : "memory");
    g += 512ull;  // 2 global rows
    l += 544u;    // 2 padded LDS rows
  }
}

__device__ __forceinline__ void stage_tile_guarded(const float* __restrict__ rf,
                                                   int tile, int E,
                                                   unsigned lds_base, int lane) {
  unsigned long long limit = (unsigned long long)E * 256ull;
#pragma unroll
  for (int i = 0; i < 8; ++i) {
    unsigned long long goff = (unsigned long long)tile * 256ull +
                              (unsigned)(lane * 16 + i * 512);
    if (goff < limit) {  // per-lane EXEC mask; garbage rows masked at scatter
      unsigned long long g = (unsigned long long)(uintptr_t)rf + goff;
      unsigned l = lds_base +
                   (unsigned)((lane >> 4) * 272 + (lane & 15) * 16 + i * 544);
      asm volatile("global_load_async_to_lds_b128 %0, %1, off"
                   :: "v"(l), "v"(g) : "memory");
    }
  }
}

// ---------------------------------------------------------------------------
// Edge kernel: one wave32 per 16-edge tile, 4 tiles per wave, double-buffered
// async LDS staging of the review_feat stream.
//   gate = sigmoid(review_feat[e,:] . ps_w)   (16 x V_WMMA_F32_16X16X4_F32)
//   out[dst,:] += gate * cj[src] * weight[src,:]   (global_atomic_add_f32)
// ---------------------------------------------------------------------------
__global__ void __launch_bounds__(256) gcmc_edge_kernel(
    const float* __restrict__ rf,      // [E,64] review_feat (HBM stream)
    const float* __restrict__ w,       // [64]   ps_w
    const float* __restrict__ weight,  // [N,64] (L2-resident gather)
    const float* __restrict__ cj,      // [N]
    const int* __restrict__ src,       // [E]
    const int* __restrict__ dst,       // [E]
    float* __restrict__ out,           // [N,64] accumulator (L2 atomics)
    int E) {
  __shared__ float stage[WAVES_PER_BLOCK][2][TILE_F];

  const int lane = threadIdx.x & 31;
  const int wib  = threadIdx.x >> 5;
  const int gwave = blockIdx.x * WAVES_PER_BLOCK + wib;
  const int gws = __builtin_amdgcn_readfirstlane(gwave);  // SGPR wave id
  const int base_e = gws * (16 * TILES_PER_WAVE);
  if (base_e >= E) return;

  const int half = lane >> 4;  // 16-lane half of the wave
  const int m    = lane & 15;  // A-matrix row (edge within tile)

  // B fragments for V_WMMA_F32_16X16X4_F32 (ps_w replicated over N columns):
  // VGPR r of B holds rows r / r+2 across the two lane halves.
  v2f bfrag[16];
#pragma unroll
  for (int c = 0; c < 16; ++c) bfrag[c] = *(const v2f*)(w + 4 * c + half * 2);

  const unsigned lds0 = (unsigned)(uintptr_t)&stage[wib][0][0];
  const unsigned lds1 = (unsigned)(uintptr_t)&stage[wib][1][0];
  const bool fullwave = (base_e + 16 * TILES_PER_WAVE) <= E;  // wave-uniform

  if (fullwave) stage_tile_fast(rf, base_e, lds0, lane);
  else          stage_tile_guarded(rf, base_e, E, lds0, lane);

#pragma unroll 1
  for (int t = 0; t < TILES_PER_WAVE; ++t) {
    const int tile = base_e + t * 16;
    if (tile >= E) break;  // wave-uniform: EXEC all-1s for WMMA

    const unsigned cur = (t & 1) ? lds1 : lds0;
    const float* sp = (const float*)(uintptr_t)((t & 1) ? &stage[wib][1][0]
                                                        : &stage[wib][0][0]);
    // Issue next tile's stage before consuming this one (double buffer).
    if (fullwave) {
      if (t + 1 < TILES_PER_WAVE) {
        stage_tile_fast(rf, tile + 16, (t & 1) ? lds0 : lds1, lane);
        asm volatile("s_wait_asynccnt 0x8" ::: "memory");  // tile t's 8 done
      } else {
        asm volatile("s_wait_asynccnt 0x0" ::: "memory");
      }
    } else {
      asm volatile("s_wait_asynccnt 0x0" ::: "memory");
      if (t + 1 < TILES_PER_WAVE && tile + 16 < E)
        stage_tile_guarded(rf, tile + 16, E, (t & 1) ? lds0 : lds1, lane);
    }

    // A-matrix 16x4 f32: lanes 0-15 hold {K0,K1}, lanes 16-31 hold {K2,K3}.
    // Padded pitch => the 32 ds_load_b64 lanes hit 64 distinct banks.
    const float* arow = sp + m * ROW_PAD_F + half * 2;
    v8f acc = {};
#pragma unroll
    for (int c = 0; c < 16; ++c) {
      v2f a = *(const v2f*)(arow + 4 * c);
      acc = __builtin_amdgcn_wmma_f32_16x16x4_f32(
          false, a, false, bfrag[c], (short)0, acc, false, false);
    }
    // D layout: each half's 16 lanes all hold that half's 8 edge-dots
    // (B column-replicated). Compute 8 sigmoids locally, swap halves once.
    float ga[8], gb[8];  // gates for edges tile+0..7 / tile+8..15
#pragma unroll
    for (int r = 0; r < 8; ++r) {
      float own = __builtin_amdgcn_rcpf(1.0f + __expf(-acc[r]));
      float oth = __shfl_xor(own, 16, 32);
      ga[r] = half ? oth : own;
      gb[r] = half ? own : oth;
    }

    // Scatter: indices are wave-uniform per edge -> scalar loads (KMcnt),
    // gather/atomics become saddr-form; 32 lanes cover the row as float2.
#pragma unroll
    for (int j = 0; j < 16; ++j) {
      const int ej = tile + j;  // uniform
      if (ej >= E) break;
      const int sj = src[ej];          // s_load
      const int dj = dst[ej];          // s_load
      const float cjs = cj[sj];        // s_load
      const float gate = (j < 8) ? ga[j] : gb[j - 8];
      const float scale = gate * cjs;
      v2f fv = *(const v2f*)(weight + (size_t)sj * D_FEAT + lane * 2);
      float* op = out + (size_t)dj * D_FEAT + lane * 2;
      unsafeAtomicAdd(op,     scale * fv.x);
      unsafeAtomicAdd(op + 1, scale * fv.y);
    }
  }
}

// ---------------------------------------------------------------------------
// Final dst-side degree normalization: out[n,:] *= ci[n] (L2-resident, b128).
// ---------------------------------------------------------------------------
__global__ void __launch_bounds__(256) gcmc_scale_kernel(
    v4f* __restrict__ out, const float* __restrict__ ci, int n4) {
  int i = blockIdx.x * blockDim.x + threadIdx.x;
  if (i < n4) {
    float c = ci[i >> 4];  // 16 float4 per 64-wide row
    v4f v = out[i];
    v.x *= c; v.y *= c; v.z *= c; v.w *= c;
    out[i] = v;
  }
}

extern "C" void kernel_launch(void* const* d_in, const int* in_sizes, int n_in,
                              void* d_out, int out_size, void* d_ws, size_t ws_size,
                              hipStream_t stream) {
  const float* weight = (const float*)d_in[0];  // [N,64]
  const float* ps_w   = (const float*)d_in[1];  // [1,64]
  const float* rf     = (const float*)d_in[2];  // [E,64]
  const float* cj     = (const float*)d_in[3];  // [N,1]
  const float* ci     = (const float*)d_in[4];  // [N,1]
  const int*   src    = (const int*)d_in[5];    // [E]
  const int*   dst    = (const int*)d_in[6];    // [E]
  float* out = (float*)d_out;

  const int E  = in_sizes[5];
  const int n4 = out_size / 4;  // out_size = N*64, multiple of 4

  gcmc_zero_kernel<<<(n4 + 255) / 256, 256, 0, stream>>>((v4f*)out, n4);

  const int edges_per_block = 16 * TILES_PER_WAVE * WAVES_PER_BLOCK;  // 512
  const int blocks = (E + edges_per_block - 1) / edges_per_block;
  gcmc_edge_kernel<<<blocks, 256, 0, stream>>>(rf, ps_w, weight, cj, src, dst,
                                               out, E);

  gcmc_scale_kernel<<<(n4 + 255) / 256, 256, 0, stream>>>((v4f*)out, ci, n4);
}